// EvalEig_42838003810955
// MI455X (gfx1250) — compile-verified
//
#include <hip/hip_runtime.h>
#include <stdint.h>

// Problem constants (from reference)
#define RN 4000
#define HSTEP 0.01f        // R_MAX/RN
#define K2 8.3333333333333333e-6f   // H*H/12

typedef __attribute__((ext_vector_type(4))) unsigned int v4u;
typedef __attribute__((ext_vector_type(8))) int         v8i;
typedef __attribute__((ext_vector_type(4))) int         v4i;

// ---------------------------------------------------------------------------
// TDM: DMA one row of `nelem` f32 from global into LDS (CDNA5 ISA).
// Verified last round: lowers to tensor_load_to_lds + s_wait_tensorcnt.
// ---------------------------------------------------------------------------
__device__ __forceinline__ void tdm_load_row_f32(const float* gsrc,
                                                 unsigned lds_byte_off,
                                                 int nelem) {
    uint64_t ga = (uint64_t)(uintptr_t)gsrc;
    v4u g0;
    g0.x = 1u;                                              // count=1 (valid user D#)
    g0.y = lds_byte_off;                                    // lds_addr (bytes)
    g0.z = (unsigned)(ga & 0xFFFFFFFFu);                    // global_addr[31:0]
    g0.w = (unsigned)((ga >> 32) & 0x01FFFFFFu)             // global_addr[56:32]
         | 0x80000000u;                                     // type=2 ("image")

    v8i g1;
    g1[0] = (int)(2u << 16);                 // wg_mask=0, data_size=2 (4 bytes)
    g1[1] = (int)((unsigned)nelem << 16);    // tensor_dim0[15:0] @ bits 63:48
    g1[2] = (int)(1u << 16);                 // tensor_dim1=1
    g1[3] = (int)((unsigned)nelem << 16);    // tile_dim0 @ bits 127:112
    g1[4] = 1;                               // tile_dim1=1, tile_dim2=0
    g1[5] = nelem;                           // tensor_dim0_stride[31:0]
    g1[6] = 0;
    g1[7] = 0;

    v4i g2 = {1, 1, 0, 0};                   // tensor_dim2=1, tensor_dim3=1
    v4i g3 = {0, (int)(1u << 16), 0, 0};     // tensor_dim4=1, tile_dim4=0

#if defined(__clang_major__) && (__clang_major__ >= 23)
    v8i g4 = {0, 0, 0, 0, 0, 0, 0, 0};
    __builtin_amdgcn_tensor_load_to_lds(g0, g1, g2, g3, g4, 0);
#else
    __builtin_amdgcn_tensor_load_to_lds(g0, g1, g2, g3, 0);
#endif
}

// sinh-series initializer: r * (1 + x^2/6 + x^4/120), x = r*sqrt(|e|)
__device__ __forceinline__ float u_inf_init(float r, float sae) {
    float x  = r * sae;
    float x2 = x * x;
    return r * (1.0f + x2 * (1.0f / 6.0f) + (x2 * x2) * (1.0f / 120.0f));
}

// One Numerov step: u_new = ((12-10*cc)*uc - cp*up) * rcp(cn)
// rcp is off the u-critical-path (cn depends only on LDS data + e).
__device__ __forceinline__ float nstep(float cp, float cc, float cn,
                                       float up, float uc) {
    return fmaf(fmaf(-10.0f, cc, 12.0f), uc, -cp * up) * __builtin_amdgcn_rcpf(cn);
}

__global__ __launch_bounds__(128) void
eval_eig_kernel(const float* __restrict__ ptl,
                const float* __restrict__ energy,
                float* __restrict__ out, int en) {
    __shared__ __align__(16) float lv[RN];   // 16 KB of the 320 KB WGP LDS pool
    const float4* lv4 = (const float4*)lv;

    // Stage potential via the Tensor Data Mover (wave 0 issues; TDM ignores EXEC).
    if (threadIdx.x < 32u) {
        tdm_load_row_f32(ptl, (unsigned)(uintptr_t)&lv[0], RN);
        __builtin_amdgcn_s_wait_tensorcnt(0);
    }
    __syncthreads();

    const int gid = (int)(blockIdx.x * blockDim.x + threadIdx.x);
    if (gid >= en) return;

    const float e  = energy[gid];
    const float ec = fmaf(K2, e, 1.0f);       // c[i] = fma(-K2, v[i], ec)

    // ---------------- forward (u_zero): only indices 0..400 matter ---------
    float r0 = HSTEP, r1 = 2.0f * HSTEP;
    float u_m2 = r0 - 0.5f * r0 * r0;         // u[0] = r - r^2/2 (l=0, PARA1=1)
    float u_m1 = r1 - 0.5f * r1 * r1;         // u[1]
    float c_m2 = fmaf(-K2, lv[0], ec);
    float c_m1 = fmaf(-K2, lv[1], ec);

    float s14 = u_m2 * u_m2;                  // phase-0 squared sum (i=0 seeded)
    float s32 = u_m1 * u_m1;                  // phase-1/3 squared sum (i=1 seeded)
    float s12 = 0.0f;                         // phase-2 LINEAR sum (reference quirk)
    const float u_in0 = u_m2;                 // endpoint u[0]

    // prologue: i=2 (phase 2), i=3 (phase 3)
    {
        float c2 = fmaf(-K2, lv[2], ec);
        float u2 = nstep(c_m2, c_m1, c2, u_m2, u_m1);
        s12 += u2;
        float c3 = fmaf(-K2, lv[3], ec);
        float u3 = nstep(c_m1, c2, c3, u_m1, u2);
        s32 += u3 * u3;
        u_m2 = u2; u_m1 = u3; c_m2 = c2; c_m1 = c3;
    }
    float a396 = 0.f, a397 = 0.f, a398 = 0.f, a399 = 0.f;
    for (int g = 1; g <= 99; ++g) {           // i = 4g .. 4g+3  (4..399)
        float4 v = lv4[g];
        float cA = fmaf(-K2, v.x, ec); float uA = nstep(c_m2, c_m1, cA, u_m2, u_m1);
        float cB = fmaf(-K2, v.y, ec); float uB = nstep(c_m1, cA, cB, u_m1, uA);
        float cC = fmaf(-K2, v.z, ec); float uC = nstep(cA, cB, cC, uA, uB);
        float cD = fmaf(-K2, v.w, ec); float uD = nstep(cB, cC, cD, uB, uC);
        if (g <= 98) {
            s14 += uA * uA;
            s32 += uB * uB + uD * uD;
            s12 += uC;
        } else {                               // g==99: i=396..399; only 396 integrates
            s14 += uA * uA;
            a396 = uA; a397 = uB; a398 = uC; a399 = uD;
        }
        u_m2 = uC; u_m1 = uD; c_m2 = cC; c_m1 = cD;
    }
    // epilogue i=400 (match point, not integrated)
    float c400 = fmaf(-K2, lv[400], ec);
    float a400 = nstep(c_m2, c_m1, c400, u_m2, u_m1);

    float integ_in = (14.0f * s14 - 7.0f * (u_in0 * u_in0 + a396 * a396)
                      + 32.0f * s32 + 12.0f * s12) * (2.0f * HSTEP / 45.0f);
    float d_in  = (25.0f * a400 - 48.0f * a399 + 36.0f * a398
                   - 16.0f * a397 + 3.0f * a396) / (12.0f * HSTEP);
    float lf_in = d_in / a400;

    // ---------------- backward (u_infty): 3999 -> 0 ------------------------
    const float sae = sqrtf(fabsf(e));
    float u_ip2 = u_inf_init(40.00f, sae);    // u[3999]
    float u_ip1 = u_inf_init(39.99f, sae);    // u[3998]
    float c_ip2 = fmaf(-K2, lv[RN - 1], ec);
    float c_ip1 = fmaf(-K2, lv[RN - 2], ec);

    float t14 = 0.f, t32 = 0.f, t12 = 0.f;
    float b396 = 0.f, b397 = 0.f, b398 = 0.f, b399 = 0.f, b400 = 0.f, b3596 = 0.f;

    // prologue: i=3997, i=3996 (outside integration range)
    {
        float c7 = fmaf(-K2, lv[3997], ec);
        float u7 = nstep(c_ip2, c_ip1, c7, u_ip2, u_ip1);
        float c6 = fmaf(-K2, lv[3996], ec);
        float u6 = nstep(c_ip1, c7, c6, u_ip1, u7);
        u_ip2 = u7; u_ip1 = u6; c_ip2 = c7; c_ip1 = c6;
    }
    for (int g = 998; g >= 0; --g) {          // i = 4g+3 .. 4g (3995..0)
        float4 v = lv4[g];
        float c3 = fmaf(-K2, v.w, ec); float u3 = nstep(c_ip2, c_ip1, c3, u_ip2, u_ip1);
        float c2 = fmaf(-K2, v.z, ec); float u2 = nstep(c_ip1, c3, c2, u_ip1, u3);
        float c1 = fmaf(-K2, v.y, ec); float u1 = nstep(c3, c2, c1, u3, u2);
        float c0 = fmaf(-K2, v.x, ec); float u0 = nstep(c2, c1, c0, u2, u1);
        if (g <= 898) {                        // all four indices <= 3595 < 3596
            t32 += u3 * u3 + u1 * u1;
            t12 += u2;
            t14 += u0 * u0;
        } else if (g == 899) {                 // only i=3596 integrates
            t14 += u0 * u0;
            b3596 = u0;
        }
        if (g == 100) b400 = u0;               // i=400
        if (g == 99) { b399 = u3; b398 = u2; b397 = u1; b396 = u0; }
        u_ip2 = u1; u_ip1 = u0; c_ip2 = c1; c_ip1 = c0;
    }
    const float u_out0 = u_ip1;               // u_infty[0]
    float integ_out = (14.0f * t14 - 7.0f * (u_out0 * u_out0 + b3596 * b3596)
                       + 32.0f * t32 + 12.0f * t12) * (2.0f * HSTEP / 45.0f);
    float d_out  = (25.0f * b400 - 48.0f * b399 + 36.0f * b398
                    - 16.0f * b397 + 3.0f * b396) / (12.0f * HSTEP);
    float lf_out = d_out / b400;

    float denom = integ_in / (a400 * a400) + integ_out / (b400 * b400);
    out[gid] = -(lf_out - lf_in) / denom;
}

extern "C" void kernel_launch(void* const* d_in, const int* in_sizes, int n_in,
                              void* d_out, int out_size, void* d_ws, size_t ws_size,
                              hipStream_t stream) {
    const float* ptl    = (const float*)d_in[0];   // RN floats
    const float* energy = (const float*)d_in[1];   // E_N floats
    float* out = (float*)d_out;                    // E_N floats (E_N x 1)
    const int en = in_sizes[1];
    const int block = 128;                         // 4 wave32 waves / workgroup
    const int grid  = (en + block - 1) / block;    // 64 workgroups for E_N=8192
    eval_eig_kernel<<<grid, block, 0, stream>>>(ptl, energy, out, en);
}